// QNNModel_3307124818536
// MI455X (gfx1250) — compile-verified
//
#include <hip/hip_runtime.h>

#define NQ 10
#define DIM 1024
#define NLAYERS 4

typedef __attribute__((ext_vector_type(2))) float v2f;
typedef __attribute__((ext_vector_type(8))) float v8f;

// Composition of the 9 CNOT permutations (control i, target i+1, i=0..8):
// src index feeding dest d after the whole ladder.
__device__ __forceinline__ int cnot_src(int d) {
#pragma unroll
  for (int i = 8; i >= 0; --i)
    if ((d >> i) & 1) d ^= (2 << i);
  return d;
}

// Build B fragment: B[k][n] = U[n][k], U = kron of 4 RY(theta) 2x2 blocks.
// Layout: vgpr v, lane L: K = 2*(L>>4)+v (+4*ks), N = L&15.
__device__ __forceinline__ void build_bfrag(const float* cc, const float* ss,
                                            int lane, v2f bfrag[4]) {
  const int half = lane >> 4;
  const int n = lane & 15;
#pragma unroll
  for (int ks = 0; ks < 4; ++ks) {
    int k0 = 4 * ks + 2 * half;
    int k1 = k0 + 1;
    float e0 = 1.f, e1 = 1.f;
#pragma unroll
    for (int q = 0; q < 4; ++q) {
      float c = cc[q], s = ss[q];
      int nb = (n >> q) & 1;
      int k0b = (k0 >> q) & 1;
      int k1b = (k1 >> q) & 1;
      e0 *= (nb == k0b) ? c : (nb ? s : -s);   // R[n][k]: [[c,-s],[s,c]]
      e1 *= (nb == k1b) ? c : (nb ? s : -s);
    }
    v2f bv; bv.x = e0; bv.y = e1;
    bfrag[ks] = bv;
  }
}

// One 1024-amplitude tensor contraction over a 4-bit index group.
// group 0: contract bits 0-3  (state as 64x16, rows = d>>4,         cols = d&15)
// group 1: contract bits 4-7  (rows = (d>>8)*16 + (d&15),           cols = (d>>4)&15)
// All A fragments are loaded before any store -> in-place safe (per-wave
// LDS ordering), which also makes folding the CNOT permutation into the
// reads legal even though it crosses tiles.
__device__ __forceinline__ void contract_group(float* S, const v2f bfrag[4],
                                               int group, bool permute, int lane) {
  const int half = lane >> 4;
  const int lid = lane & 15;
  v2f a[16];
#pragma unroll
  for (int t = 0; t < 4; ++t) {
#pragma unroll
    for (int ks = 0; ks < 4; ++ks) {
      int k0 = 4 * ks + 2 * half;
      int d0, d1;
      if (group == 0) { int m = t * 16 + lid; d0 = (m << 4) | k0; d1 = d0 + 1; }
      else            { d0 = t * 256 + k0 * 16 + lid; d1 = d0 + 16; }
      if (permute) { d0 = cnot_src(d0); d1 = cnot_src(d1); }
      v2f av; av.x = S[d0]; av.y = S[d1];
      a[t * 4 + ks] = av;
    }
  }
#pragma unroll
  for (int t = 0; t < 4; ++t) {
    v8f acc = {0.f, 0.f, 0.f, 0.f, 0.f, 0.f, 0.f, 0.f};
#pragma unroll
    for (int ks = 0; ks < 4; ++ks)
      acc = __builtin_amdgcn_wmma_f32_16x16x4_f32(
          false, a[t * 4 + ks], false, bfrag[ks], (short)0, acc, false, false);
#pragma unroll
    for (int r = 0; r < 8; ++r) {
      int m = r + 8 * half;                       // C/D layout: M = r + 8*half, N = lid
      int d = (group == 0) ? (((t * 16 + m) << 4) | lid)
                           : (t * 256 + lid * 16 + m);
      S[d] = acc[r];
    }
  }
}

// RY on qubits 8 and 9: 4x4 contraction over the top 2 bits, scalar.
__device__ __forceinline__ void contract_top(float* Sre, float* Sim,
                                             float c8, float s8,
                                             float c9, float s9, int lane) {
#pragma unroll
  for (int j = 0; j < 8; ++j) {
    int rest = j * 32 + lane;                     // each lane owns its rest values
    {
      float a0 = Sre[rest], a1 = Sre[256 + rest];
      float a2 = Sre[512 + rest], a3 = Sre[768 + rest];
      float t0 = c8 * a0 - s8 * a1, t1 = s8 * a0 + c8 * a1;
      float t2 = c8 * a2 - s8 * a3, t3 = s8 * a2 + c8 * a3;
      Sre[rest]       = c9 * t0 - s9 * t2;
      Sre[256 + rest] = c9 * t1 - s9 * t3;
      Sre[512 + rest] = s9 * t0 + c9 * t2;
      Sre[768 + rest] = s9 * t1 + c9 * t3;
    }
    {
      float a0 = Sim[rest], a1 = Sim[256 + rest];
      float a2 = Sim[512 + rest], a3 = Sim[768 + rest];
      float t0 = c8 * a0 - s8 * a1, t1 = s8 * a0 + c8 * a1;
      float t2 = c8 * a2 - s8 * a3, t3 = s8 * a2 + c8 * a3;
      Sim[rest]       = c9 * t0 - s9 * t2;
      Sim[256 + rest] = c9 * t1 - s9 * t3;
      Sim[512 + rest] = s9 * t0 + c9 * t2;
      Sim[768 + rest] = s9 * t1 + c9 * t3;
    }
  }
}

__global__ __launch_bounds__(32) void qnn_circuit_kernel(
    const float* __restrict__ x, const float* __restrict__ W_pre,
    const float* __restrict__ b_pre, const float* __restrict__ weights,
    const float* __restrict__ W_post, const float* __restrict__ b_post,
    float* __restrict__ out) {
  __shared__ float Sre[DIM];
  __shared__ float Sim[DIM];
  __shared__ float hs[16];
  __shared__ float gs[16];
  __shared__ float csc[(NLAYERS + 1) * NQ];
  __shared__ float css[(NLAYERS + 1) * NQ];
  __shared__ float red[6 * 32];
  __shared__ float zb[16];

  const int lane = threadIdx.x;   // one wave32 per batch element
  const int b = blockIdx.x;

  // cos/sin of half-angles for all 50 RY gates
  for (int idx = lane; idx < (NLAYERS + 1) * NQ; idx += 32) {
    float w = 0.5f * weights[idx];
    csc[idx] = __cosf(w);
    css[idx] = __sinf(w);
  }
  // h = x @ W_pre^T + b_pre  (10x10, lanes 0..9)
  if (lane < NQ) {
    float acc = b_pre[lane];
#pragma unroll
    for (int k = 0; k < NQ; ++k) acc += x[b * NQ + k] * W_pre[lane * NQ + k];
    hs[lane] = acc;
  }
  __syncthreads();
  if (lane < NQ - 1) {
    const float PI = 3.14159265358979f;
    gs[lane] = (PI - hs[lane]) * (PI - hs[lane + 1]);
  }
  __syncthreads();

  float hr[NQ], gr[NQ - 1];
#pragma unroll
  for (int q = 0; q < NQ; ++q) hr[q] = hs[q];
#pragma unroll
  for (int q = 0; q < NQ - 1; ++q) gr[q] = gs[q];

  // initial state: (1/32) * exp(-i*ang(d))
  const float inv = 0.03125f;
#pragma unroll 4
  for (int j = 0; j < 32; ++j) {
    int d = (j << 5) | lane;                    // lane in low bits: conflict-free banks
    float ang = 0.f;
#pragma unroll
    for (int q = 0; q < NQ; ++q) ang += ((d >> q) & 1) ? -hr[q] : hr[q];
    int dz = d ^ (d >> 1);                      // zz sign = parity of adjacent bits
#pragma unroll
    for (int q = 0; q < NQ - 1; ++q) ang += ((dz >> q) & 1) ? -gr[q] : gr[q];
    float sn, cn;
    __sincosf(ang, &sn, &cn);
    Sre[d] = cn * inv;
    Sim[d] = -sn * inv;
  }
  __syncthreads();

  // 5 RY layers; CNOT ladder of layer l-1 folded into layer l's low-group reads
  v2f bfrag[4];
#pragma unroll 1
  for (int l = 0; l <= NLAYERS; ++l) {
    build_bfrag(&csc[l * NQ + 0], &css[l * NQ + 0], lane, bfrag);   // qubits 0..3
    contract_group(Sre, bfrag, 0, l > 0, lane);
    contract_group(Sim, bfrag, 0, l > 0, lane);
    __syncthreads();
    build_bfrag(&csc[l * NQ + 4], &css[l * NQ + 4], lane, bfrag);   // qubits 4..7
    contract_group(Sre, bfrag, 1, false, lane);
    contract_group(Sim, bfrag, 1, false, lane);
    __syncthreads();
    contract_top(Sre, Sim, csc[l * NQ + 8], css[l * NQ + 8],        // qubits 8,9
                 csc[l * NQ + 9], css[l * NQ + 9], lane);
    __syncthreads();
  }

  // readout: z_q = sum_d |amp(d)|^2 * (1 - 2*bit_q(d));  d = j*32 + lane
  float psum = 0.f, zh0 = 0.f, zh1 = 0.f, zh2 = 0.f, zh3 = 0.f, zh4 = 0.f;
#pragma unroll 4
  for (int j = 0; j < 32; ++j) {
    int d = (j << 5) | lane;
    float re = Sre[d], im = Sim[d];
    float p = re * re + im * im;
    psum += p;
    zh0 += (j & 1) ? -p : p;
    zh1 += (j & 2) ? -p : p;
    zh2 += (j & 4) ? -p : p;
    zh3 += (j & 8) ? -p : p;
    zh4 += (j & 16) ? -p : p;
  }
  red[lane] = psum;
  red[32 + lane] = zh0;  red[64 + lane] = zh1;  red[96 + lane] = zh2;
  red[128 + lane] = zh3; red[160 + lane] = zh4;
  __syncthreads();
  if (lane < NQ) {
    float z = 0.f;
    if (lane < 5) {
#pragma unroll
      for (int L = 0; L < 32; ++L) z += ((L >> lane) & 1) ? -red[L] : red[L];
    } else {
#pragma unroll
      for (int L = 0; L < 32; ++L) z += red[(lane - 4) * 32 + L];
    }
    zb[lane] = z;
  }
  __syncthreads();
  if (lane < NQ) {
    float acc = b_post[lane];
#pragma unroll
    for (int q = 0; q < NQ; ++q) acc += zb[q] * W_post[lane * NQ + q];
    out[b * NQ + lane] = acc;
  }
}

extern "C" void kernel_launch(void* const* d_in, const int* in_sizes, int n_in,
                              void* d_out, int out_size, void* d_ws, size_t ws_size,
                              hipStream_t stream) {
  const float* x       = (const float*)d_in[0];
  const float* W_pre   = (const float*)d_in[1];
  const float* b_pre   = (const float*)d_in[2];
  const float* weights = (const float*)d_in[3];
  const float* W_post  = (const float*)d_in[4];
  const float* b_post  = (const float*)d_in[5];
  float* out = (float*)d_out;
  const int batch = in_sizes[0] / NQ;   // 2048
  hipLaunchKernelGGL(qnn_circuit_kernel, dim3(batch), dim3(32), 0, stream,
                     x, W_pre, b_pre, weights, W_post, b_post, out);
}